// GNNWithAttention_6665789244019
// MI455X (gfx1250) — compile-verified
//
#include <hip/hip_runtime.h>
#include <hip/hip_bf16.h>
#include <stdint.h>

// ---------------------------------------------------------------------------
// GAT pipeline for MI455X (gfx1250, wave32).
// GEMMs use V_WMMA_F32_16X16X4_F32 (exact fp32). Edge phase is L2-atomic
// streaming (hW fits in 192MB L2; 23.3 TB/s HBM bound ~150us/conv).
// ---------------------------------------------------------------------------

typedef __attribute__((ext_vector_type(2))) float v2f;
typedef __attribute__((ext_vector_type(8))) float v8f;

#define HIDC 128
#define HEADS 4
#define DHEAD 32
#define NEG_SLOPE 0.2f
#define LN_EPS 1e-5f

// ---------------- generic fp32 WMMA GEMM: C[M,Nc] = A[M,K] @ B[K,Nc] (+bias)
// One wave computes one 16x16 output tile. 4 waves/block cover 64 cols.
// A-matrix 16x4 lane layout (ISA 7.12.2): lane l -> m=l%16,
//   VGPR0 = A[m][k + 2*(l/16)], VGPR1 = A[m][k + 2*(l/16)+1].
// B-matrix 4x16: lane l -> n=l%16, VGPR0 = B[k+2*(l/16)][n], VGPR1 = next k.
// C/D: VGPR r, lane l -> M = r + 8*(l/16), N = l%16.
__global__ __launch_bounds__(128) void gemm_wmma_f32(
    const float* __restrict__ A, const float* __restrict__ B,
    const float* __restrict__ bias, float* __restrict__ C,
    int M, int K, int Ncols)
{
    const int wave = threadIdx.x >> 5;
    const int lane = threadIdx.x & 31;
    const int half = lane >> 4;
    const int l16  = lane & 15;

    const int ntile = blockIdx.x * 4 + wave;   // wave-uniform
    const int mtile = blockIdx.y;
    if (ntile * 16 >= Ncols) return;           // wave-uniform exit (EXEC stays full)
    if (mtile * 16 >= M) return;

    const float* Arow  = A + (size_t)(mtile * 16 + l16) * K;
    const float* Bcol  = B + (size_t)(ntile * 16 + l16);

    v8f acc = {};
    for (int k = 0; k < K; k += 4) {
        v2f a, b;
        a.x = Arow[k + 2 * half];
        a.y = Arow[k + 2 * half + 1];
        b.x = Bcol[(size_t)(k + 2 * half) * Ncols];
        b.y = Bcol[(size_t)(k + 2 * half + 1) * Ncols];
        acc = __builtin_amdgcn_wmma_f32_16x16x4_f32(
            false, a, false, b, (short)0, acc, false, false);
    }

    const float bval = bias ? bias[ntile * 16 + l16] : 0.0f;
    #pragma unroll
    for (int r = 0; r < 8; ++r) {
        const int row = mtile * 16 + r + 8 * half;
        C[(size_t)row * Ncols + ntile * 16 + l16] = acc[r] + bval;
    }
}

// ---------------- fused LayerNorm + GELU(tanh approx), one wave per 128-row
__global__ __launch_bounds__(128) void ln_gelu_kernel(
    const float* __restrict__ in, const float* __restrict__ g,
    const float* __restrict__ be, float* __restrict__ out, int N)
{
    const int lane = threadIdx.x & 31;
    const int row  = blockIdx.x * 4 + (threadIdx.x >> 5);
    if (row >= N) return;

    float4 v = ((const float4*)(in + (size_t)row * HIDC))[lane];

    float s = v.x + v.y + v.z + v.w;
    #pragma unroll
    for (int m = 16; m >= 1; m >>= 1) s += __shfl_xor(s, m, 32);
    const float mu = s * (1.0f / HIDC);

    const float d0 = v.x - mu, d1 = v.y - mu, d2 = v.z - mu, d3 = v.w - mu;
    float q = d0 * d0 + d1 * d1 + d2 * d2 + d3 * d3;
    #pragma unroll
    for (int m = 16; m >= 1; m >>= 1) q += __shfl_xor(q, m, 32);
    const float inv = rsqrtf(q * (1.0f / HIDC) + LN_EPS);

    float4 gg = ((const float4*)g)[lane];
    float4 bb = ((const float4*)be)[lane];

    float y[4];
    y[0] = d0 * inv * gg.x + bb.x;
    y[1] = d1 * inv * gg.y + bb.y;
    y[2] = d2 * inv * gg.z + bb.z;
    y[3] = d3 * inv * gg.w + bb.w;

    #pragma unroll
    for (int j = 0; j < 4; ++j) {
        const float x = y[j];
        const float c = 0.7978845608028654f * (x + 0.044715f * x * x * x);
        y[j] = 0.5f * x * (1.0f + tanhf(c));
    }
    float4 o = make_float4(y[0], y[1], y[2], y[3]);
    ((float4*)(out + (size_t)row * HIDC))[lane] = o;
}

// ---------------- per-node attention logits: e = sum_d hW[n,h,d] * a[h,d]
// one wave per node; lanes in groups of 8 cover one head (8*4 = 32 = D)
__global__ __launch_bounds__(128) void attn_score_kernel(
    const float* __restrict__ hW, const float* __restrict__ a_src,
    const float* __restrict__ a_dst, float* __restrict__ esrc,
    float* __restrict__ edst, int N)
{
    const int lane = threadIdx.x & 31;
    const int row  = blockIdx.x * 4 + (threadIdx.x >> 5);
    if (row >= N) return;

    float4 v  = ((const float4*)(hW + (size_t)row * HIDC))[lane];
    float4 as = ((const float4*)a_src)[lane];
    float4 ad = ((const float4*)a_dst)[lane];

    float ps = v.x * as.x + v.y * as.y + v.z * as.z + v.w * as.w;
    float pd = v.x * ad.x + v.y * ad.y + v.z * ad.z + v.w * ad.w;
    #pragma unroll
    for (int m = 1; m <= 4; m <<= 1) {
        ps += __shfl_xor(ps, m, 32);
        pd += __shfl_xor(pd, m, 32);
    }
    if ((lane & 7) == 0) {
        esrc[(size_t)row * HEADS + (lane >> 3)] = ps;
        edst[(size_t)row * HEADS + (lane >> 3)] = pd;
    }
}

// ---------------- fills
__global__ void fill_val_kernel(float* __restrict__ p, float val, long long n) {
    long long i = blockIdx.x * (long long)blockDim.x + threadIdx.x;
    if (i < n) p[i] = val;
}
__global__ void fill_bias_rows_kernel(float* __restrict__ out,
                                      const float* __restrict__ bias, int N) {
    long long i = blockIdx.x * (long long)blockDim.x + threadIdx.x;
    if (i < (long long)N * HIDC) out[i] = bias[i & (HIDC - 1)];
}

// ordered-int float atomic max: maps to native GLOBAL_ATOMIC_MAX_I32/MIN_U32
__device__ __forceinline__ void atomicMaxF(float* addr, float val) {
    if (val >= 0.0f) atomicMax((int*)addr, __float_as_int(val));
    else             atomicMin((unsigned int*)addr, __float_as_uint(val));
}

__device__ __forceinline__ float leaky(float s) {
    return s > 0.0f ? s : NEG_SLOPE * s;
}

// ---------------- edge pass A: per-dst-per-head running max of logits
__global__ __launch_bounds__(256) void edge_max_kernel(
    const int* __restrict__ src_idx, const int* __restrict__ dst_idx,
    const float* __restrict__ esrc, const float* __restrict__ edst,
    float* __restrict__ mmax, long long E, int N)
{
    const long long e = blockIdx.x * (long long)blockDim.x + threadIdx.x;
    if (e >= E + (long long)N) return;
    const int s = (e < E) ? src_idx[e] : (int)(e - E);
    const int d = (e < E) ? dst_idx[e] : (int)(e - E);
    #pragma unroll
    for (int h = 0; h < HEADS; ++h) {
        const float sc = leaky(esrc[(size_t)s * HEADS + h] + edst[(size_t)d * HEADS + h]);
        atomicMaxF(&mmax[(size_t)d * HEADS + h], sc);
    }
}

// ---------------- edge pass B: softmax denominator
__global__ __launch_bounds__(256) void edge_sum_kernel(
    const int* __restrict__ src_idx, const int* __restrict__ dst_idx,
    const float* __restrict__ esrc, const float* __restrict__ edst,
    const float* __restrict__ mmax, float* __restrict__ denom,
    long long E, int N)
{
    const long long e = blockIdx.x * (long long)blockDim.x + threadIdx.x;
    if (e >= E + (long long)N) return;
    const int s = (e < E) ? src_idx[e] : (int)(e - E);
    const int d = (e < E) ? dst_idx[e] : (int)(e - E);
    #pragma unroll
    for (int h = 0; h < HEADS; ++h) {
        const float sc = leaky(esrc[(size_t)s * HEADS + h] + edst[(size_t)d * HEADS + h]);
        atomicAdd(&denom[(size_t)d * HEADS + h], __expf(sc - mmax[(size_t)d * HEADS + h]));
    }
}

// ---------------- edge pass C: out[dst] += alpha * hW[src]; wave per edge
__global__ __launch_bounds__(256) void edge_agg_kernel(
    const int* __restrict__ src_idx, const int* __restrict__ dst_idx,
    const float* __restrict__ esrc, const float* __restrict__ edst,
    const float* __restrict__ mmax, const float* __restrict__ denom,
    const float* __restrict__ hW, float* __restrict__ out,
    long long E, int N)
{
    const int lane = threadIdx.x & 31;
    const long long e = blockIdx.x * (long long)(blockDim.x >> 5) + (threadIdx.x >> 5);
    if (e >= E + (long long)N) return;
    const int s = (e < E) ? src_idx[e] : (int)(e - E);
    const int d = (e < E) ? dst_idx[e] : (int)(e - E);

    const int h = lane >> 3;  // 8 lanes per head
    const float sc = leaky(esrc[(size_t)s * HEADS + h] + edst[(size_t)d * HEADS + h]);
    const float alpha = __expf(sc - mmax[(size_t)d * HEADS + h]) / denom[(size_t)d * HEADS + h];

    float4 v = ((const float4*)(hW + (size_t)s * HIDC))[lane];
    float* o = out + (size_t)d * HIDC + lane * 4;
    atomicAdd(o + 0, alpha * v.x);
    atomicAdd(o + 1, alpha * v.y);
    atomicAdd(o + 2, alpha * v.z);
    atomicAdd(o + 3, alpha * v.w);
}

// ---------------------------------------------------------------------------
extern "C" void kernel_launch(void* const* d_in, const int* in_sizes, int n_in,
                              void* d_out, int out_size, void* d_ws, size_t ws_size,
                              hipStream_t stream) {
    const float* x     = (const float*)d_in[0];
    const int*   eidx  = (const int*)d_in[1];
    const float* w_in  = (const float*)d_in[2];
    const float* b_in  = (const float*)d_in[3];
    const float* g1    = (const float*)d_in[4];
    const float* be1   = (const float*)d_in[5];
    const float* W1    = (const float*)d_in[6];
    const float* as1   = (const float*)d_in[7];
    const float* ad1   = (const float*)d_in[8];
    const float* bg1   = (const float*)d_in[9];
    const float* g2    = (const float*)d_in[10];
    const float* be2   = (const float*)d_in[11];
    const float* W2    = (const float*)d_in[12];
    const float* as2   = (const float*)d_in[13];
    const float* ad2   = (const float*)d_in[14];
    const float* bg2   = (const float*)d_in[15];
    const float* w_out = (const float*)d_in[16];
    const float* b_out = (const float*)d_in[17];

    const int       N   = 50000;
    const int       IN  = 64;
    const int       OUT = 64;
    const long long E   = in_sizes[1] / 2;
    const long long TOT = E + (long long)N;
    const int* src = eidx;
    const int* dst = eidx + E;

    // workspace layout (floats)
    float* ws   = (float*)d_ws;
    float* bufA = ws;                         // N*128 (post ln/gelu)
    float* bufB = bufA + (size_t)N * HIDC;    // N*128 (hW)
    float* bufC = bufB + (size_t)N * HIDC;    // N*128 (h / aggregate out)
    float* es   = bufC + (size_t)N * HIDC;    // N*4
    float* ed   = es + (size_t)N * HEADS;     // N*4
    float* mm   = ed + (size_t)N * HEADS;     // N*4
    float* dn   = mm + (size_t)N * HEADS;     // N*4

    const dim3 blk128(128), blk256(256);
    const dim3 gRows((N + 3) / 4);                          // wave-per-row kernels
    const dim3 gEdge((unsigned)((TOT + 255) / 256));        // thread-per-edge
    const dim3 gAgg((unsigned)((TOT + 7) / 8));             // wave-per-edge
    const dim3 gNH((unsigned)(((long long)N * HEADS + 255) / 256));
    const dim3 gNC((unsigned)(((long long)N * HIDC + 255) / 256));

    // 1) h = x @ w_in + b_in
    gemm_wmma_f32<<<dim3(HIDC / 64, N / 16), blk128, 0, stream>>>(
        x, w_in, b_in, bufC, N, IN, HIDC);

    for (int layer = 0; layer < 2; ++layer) {
        const float* g  = layer ? g2  : g1;
        const float* be = layer ? be2 : be1;
        const float* W  = layer ? W2  : W1;
        const float* as = layer ? as2 : as1;
        const float* ad = layer ? ad2 : ad1;
        const float* bg = layer ? bg2 : bg1;

        // 2) ln + gelu
        ln_gelu_kernel<<<gRows, blk128, 0, stream>>>(bufC, g, be, bufA, N);
        // 3) hW = h @ W  (no bias at GEMM; bg added into aggregate init)
        gemm_wmma_f32<<<dim3(HIDC / 64, N / 16), blk128, 0, stream>>>(
            bufA, W, nullptr, bufB, N, HIDC, HIDC);
        // 4) attention logits per node/head
        attn_score_kernel<<<gRows, blk128, 0, stream>>>(bufB, as, ad, es, ed, N);
        // 5) init softmax state + output-with-bias
        fill_val_kernel<<<gNH, blk256, 0, stream>>>(mm, -INFINITY, (long long)N * HEADS);
        fill_val_kernel<<<gNH, blk256, 0, stream>>>(dn, 0.0f, (long long)N * HEADS);
        fill_bias_rows_kernel<<<gNC, blk256, 0, stream>>>(bufC, bg, N);
        // 6) segment softmax + aggregation over edges (incl. self loops)
        edge_max_kernel<<<gEdge, blk256, 0, stream>>>(src, dst, es, ed, mm, E, N);
        edge_sum_kernel<<<gEdge, blk256, 0, stream>>>(src, dst, es, ed, mm, dn, E, N);
        edge_agg_kernel<<<gAgg, blk256, 0, stream>>>(src, dst, es, ed, mm, dn, bufB, bufC, E, N);
    }

    // 7) out = h @ w_out + b_out
    gemm_wmma_f32<<<dim3(OUT / 64, N / 16), blk128, 0, stream>>>(
        bufC, w_out, b_out, (float*)d_out, N, HIDC, OUT);

    (void)n_in; (void)out_size; (void)ws_size;
}